// LDGCNN_5729486372888
// MI455X (gfx1250) — compile-verified
//
#include <hip/hip_runtime.h>

typedef __attribute__((ext_vector_type(2))) float v2f;
typedef __attribute__((ext_vector_type(8))) float v8f;

#define NPTS   1024
#define BATCH  8
#define KNN    20
#define XC_CH  323
#define EPSV   1e-5f
#define NEGINF (-3.402823466e38f)

// ---------------------------------------------------------------------------
// copy x (B,3,N) into prefix channels of Xcat (B,323,N)
// ---------------------------------------------------------------------------
__global__ void k_copy_x(const float* __restrict__ x, float* __restrict__ XC) {
    int t = blockIdx.x * 256 + threadIdx.x;          // 0 .. 8*3*1024-1
    int b = t / (3 * NPTS);
    int r = t - b * 3 * NPTS;                         // c*1024 + n
    XC[(size_t)b * XC_CH * NPTS + r] = x[t];
}

// ---------------------------------------------------------------------------
// xx[b][n] = sum_c X[b][chanOff+c][n]^2
// ---------------------------------------------------------------------------
__global__ void k_xx(const float* __restrict__ XC, float* __restrict__ xx,
                     int chanOff, int C) {
    int t = blockIdx.x * 256 + threadIdx.x;          // b*1024+n
    int b = t >> 10, n = t & 1023;
    const float* Xb = XC + (size_t)b * XC_CH * NPTS + (size_t)chanOff * NPTS + n;
    float s = 0.f;
    for (int c = 0; c < C; ++c) { float v = Xb[(size_t)c * NPTS]; s += v * v; }
    xx[t] = s;
}

// ---------------------------------------------------------------------------
// dist[b][i][j] = 2*<x_i,x_j> - xx[i] - xx[j]   via V_WMMA_F32_16X16X4_F32
// wave -> 16 rows x 64 cols (4 j-subtiles, A fragment reused 4x per k-step)
// block = 4 waves stacked in i -> 64x64 block tile
// ---------------------------------------------------------------------------
__global__ __launch_bounds__(128)
void k_gram_dist(const float* __restrict__ XC, const float* __restrict__ xx,
                 float* __restrict__ dist, int chanOff, int C) {
    const int lane = threadIdx.x & 31;
    const int wave = threadIdx.x >> 5;
    const int b  = blockIdx.z;
    const int j0 = blockIdx.x * 64;                   // 4 col subtiles
    const int i0 = blockIdx.y * 64 + wave * 16;       // rows
    const int half = lane >> 4;
    const int lr   = lane & 15;
    const float* Xb = XC + (size_t)b * XC_CH * NPTS + (size_t)chanOff * NPTS;
    v8f acc[4] = {};
    for (int k0 = 0; k0 < C; k0 += 4) {
        int ka = k0 + 2 * half;                       // per-lane K slots (ISA A/B layout)
        int kb = ka + 1;
        float ma = (ka < C) ? 1.f : 0.f;
        float mb = (kb < C) ? 1.f : 0.f;
        int kac = (ka < C) ? ka : 0;
        int kbc = (kb < C) ? kb : 0;
        v2f a;
        a.x = Xb[(size_t)kac * NPTS + i0 + lr] * ma;  // A = X^T : (i, k), reused 4x
        a.y = Xb[(size_t)kbc * NPTS + i0 + lr] * mb;
#pragma unroll
        for (int jt = 0; jt < 4; ++jt) {
            v2f bm;
            bm.x = Xb[(size_t)kac * NPTS + j0 + jt * 16 + lr] * ma;
            bm.y = Xb[(size_t)kbc * NPTS + j0 + jt * 16 + lr] * mb;
            acc[jt] = __builtin_amdgcn_wmma_f32_16x16x4_f32(
                false, a, false, bm, (short)0, acc[jt], false, false);
        }
    }
    const float* xxb = xx + b * NPTS;
    float xxi[8];
#pragma unroll
    for (int r = 0; r < 8; ++r) xxi[r] = xxb[i0 + r + 8 * half];
    float* db = dist + ((size_t)b << 20);
#pragma unroll
    for (int jt = 0; jt < 4; ++jt) {
        float xxj = xxb[j0 + jt * 16 + lr];
#pragma unroll
        for (int r = 0; r < 8; ++r) {                 // C/D layout: VGPR r -> M=r / r+8
            int i = i0 + r + 8 * half;
            db[(size_t)i * NPTS + (j0 + jt * 16 + lr)] = 2.f * acc[jt][r] - xxi[r] - xxj;
        }
    }
}

// ---------------------------------------------------------------------------
// top-k (k=20) per row, ties -> lower index first (matches jax.lax.top_k)
// ---------------------------------------------------------------------------
__global__ void k_topk(const float* __restrict__ dist, int* __restrict__ idx) {
    int t = blockIdx.x * 256 + threadIdx.x;           // b*1024 + i
    int b = t >> 10, i = t & 1023;
    const float* row = dist + ((size_t)b << 20) + ((size_t)i << 10);
    float bv[KNN]; int bi[KNN];
    for (int k = 0; k < KNN; ++k) { bv[k] = NEGINF; bi[k] = 0; }
    for (int j = 0; j < NPTS; ++j) {
        float v = row[j];
        if (v > bv[KNN - 1]) {
            int p = KNN - 1;
            while (p > 0 && v > bv[p - 1]) { bv[p] = bv[p - 1]; bi[p] = bi[p - 1]; --p; }
            bv[p] = v; bi[p] = j;
        }
    }
    for (int k = 0; k < KNN; ++k) idx[(size_t)t * KNN + k] = bi[k];
}

// ---------------------------------------------------------------------------
// Wstk (2co x ci): rows [0,co)=W_a, rows [co,2co)=W_b - W_a  (w is co x 2ci)
// ---------------------------------------------------------------------------
__global__ void k_stack_w(const float* __restrict__ w, float* __restrict__ Wstk,
                          int co, int ci) {
    int t = blockIdx.x * 256 + threadIdx.x;
    if (t >= 2 * co * ci) return;
    int m = t / ci, c = t - m * ci;
    float v;
    if (m < co) v = w[(size_t)m * 2 * ci + c];
    else {
        int o = m - co;
        v = w[(size_t)o * 2 * ci + ci + c] - w[(size_t)o * 2 * ci + c];
    }
    Wstk[t] = v;
}

// ---------------------------------------------------------------------------
// Y[b][m][n] = sum_k A[m][k] * XC[b][k][n]  via V_WMMA_F32_16X16X4_F32
// wave -> 64 rows x 16 cols (4 m-subtiles, B fragment reused 4x per k-step)
// XC prefix channels (stride XC_CH*NPTS per batch); Y stride M*NPTS.
// ---------------------------------------------------------------------------
__global__ __launch_bounds__(128)
void k_gemm(const float* __restrict__ A, const float* __restrict__ XC,
            float* __restrict__ Y, int M, int K) {
    const int lane = threadIdx.x & 31;
    const int wave = threadIdx.x >> 5;
    const int b  = blockIdx.z;
    const int n0 = blockIdx.x * 64 + wave * 16;
    const int m0 = blockIdx.y * 64;                   // 4 m subtiles
    const int half = lane >> 4;
    const int lr   = lane & 15;
    const float* Xb = XC + (size_t)b * XC_CH * NPTS;
    float* Yb = Y + (size_t)b * M * NPTS;
    v8f acc[4] = {};
    for (int k0 = 0; k0 < K; k0 += 4) {
        int ka = k0 + 2 * half;
        int kb = ka + 1;
        float ma = (ka < K) ? 1.f : 0.f;
        float mb = (kb < K) ? 1.f : 0.f;
        int kac = (ka < K) ? ka : 0;
        int kbc = (kb < K) ? kb : 0;
        v2f bm;                                        // B fragment, reused 4x
        bm.x = Xb[(size_t)kac * NPTS + n0 + lr] * ma;
        bm.y = Xb[(size_t)kbc * NPTS + n0 + lr] * mb;
#pragma unroll
        for (int mt = 0; mt < 4; ++mt) {
            v2f a;
            a.x = A[(size_t)(m0 + mt * 16 + lr) * K + kac] * ma;
            a.y = A[(size_t)(m0 + mt * 16 + lr) * K + kbc] * mb;
            acc[mt] = __builtin_amdgcn_wmma_f32_16x16x4_f32(
                false, a, false, bm, (short)0, acc[mt], false, false);
        }
    }
#pragma unroll
    for (int mt = 0; mt < 4; ++mt)
#pragma unroll
        for (int r = 0; r < 8; ++r)
            Yb[(size_t)(m0 + mt * 16 + r + 8 * half) * NPTS + n0 + lr] = acc[mt][r];
}

// ---------------------------------------------------------------------------
// gather neighbors: z = Ya[b][o][idx] + Yc[b][o][n]; max over k + BN stats
// grid: (B*N/256, co); Ya layout (B, 2co, N) with Yc in rows [co,2co)
// ---------------------------------------------------------------------------
__global__ void k_edge_reduce(const float* __restrict__ Ya, const int* __restrict__ idx,
                              float* __restrict__ mbuf, float* __restrict__ sums,
                              int co) {
    int o = blockIdx.y;
    int t = blockIdx.x * 256 + threadIdx.x;           // b*1024+n
    int b = t >> 10, n = t & 1023;
    const float* YaB   = Ya + (size_t)b * 2 * co * NPTS;
    const float* yaRow = YaB + (size_t)o * NPTS;
    float yc = YaB[(size_t)(co + o) * NPTS + n];
    const int* ip = idx + (size_t)t * KNN;
    float mx = NEGINF, s = 0.f, s2 = 0.f;
#pragma unroll
    for (int k = 0; k < KNN; ++k) {
        float z = yaRow[ip[k]] + yc;
        mx = fmaxf(mx, z); s += z; s2 += z * z;
    }
    mbuf[((size_t)b * co + o) * NPTS + n] = mx;
    __shared__ float sh[2];
    if (threadIdx.x == 0) { sh[0] = 0.f; sh[1] = 0.f; }
    __syncthreads();
    atomicAdd(&sh[0], s); atomicAdd(&sh[1], s2);
    __syncthreads();
    if (threadIdx.x == 0) {
        atomicAdd(&sums[o], sh[0]);
        atomicAdd(&sums[512 + o], sh[1]);
    }
}

// BN stats for the K=1 final layer: sums over (b,n) of y, y^2
__global__ void k_stats_plain(const float* __restrict__ Y, float* __restrict__ sums,
                              int M) {
    int o = blockIdx.y;
    int t = blockIdx.x * 256 + threadIdx.x;
    int b = t >> 10, n = t & 1023;
    float y = Y[((size_t)b * M + o) * NPTS + n];
    __shared__ float sh[2];
    if (threadIdx.x == 0) { sh[0] = 0.f; sh[1] = 0.f; }
    __syncthreads();
    atomicAdd(&sh[0], y); atomicAdd(&sh[1], y * y);
    __syncthreads();
    if (threadIdx.x == 0) {
        atomicAdd(&sums[o], sh[0]);
        atomicAdd(&sums[512 + o], sh[1]);
    }
}

__global__ void k_bn_params(const float* __restrict__ sums, const float* __restrict__ g,
                            const float* __restrict__ bb, float* __restrict__ scsh,
                            int M, float invCnt) {
    int o = blockIdx.x * 256 + threadIdx.x;
    if (o >= M) return;
    float mean = sums[o] * invCnt;
    float var  = sums[512 + o] * invCnt - mean * mean;
    float sc   = g[o] * rsqrtf(var + EPSV);
    scsh[o]       = sc;
    scsh[512 + o] = bb[o] - mean * sc;
}

// BN+ReLU applied to per-point max; write into Xcat at channel outOff
__global__ void k_apply_feat(const float* __restrict__ mbuf, const float* __restrict__ scsh,
                             float* __restrict__ XC, int co, int outOff) {
    int t = blockIdx.x * 256 + threadIdx.x;           // over B*co*N
    int n = t & 1023;
    int r = t >> 10;
    int o = r % co, b = r / co;
    float v = scsh[o] * mbuf[t] + scsh[512 + o];
    XC[(size_t)b * XC_CH * NPTS + (size_t)(outOff + o) * NPTS + n] = v > 0.f ? v : 0.f;
}

// final BN+ReLU straight to output (flat (B,512,N) == src;tgt concat)
__global__ void k_apply_out(const float* __restrict__ Y, const float* __restrict__ scsh,
                            float* __restrict__ out) {
    int t = blockIdx.x * 256 + threadIdx.x;           // over 8*512*1024
    int o = (t >> 10) & 511;
    float v = scsh[o] * Y[t] + scsh[512 + o];
    out[t] = v > 0.f ? v : 0.f;
}

// ---------------------------------------------------------------------------
extern "C" void kernel_launch(void* const* d_in, const int* in_sizes, int n_in,
                              void* d_out, int out_size, void* d_ws, size_t ws_size,
                              hipStream_t stream) {
    (void)in_sizes; (void)n_in; (void)out_size; (void)ws_size;
    const float* x  = (const float*)d_in[0];
    const float* w[5]  = {(const float*)d_in[1], (const float*)d_in[4], (const float*)d_in[7],
                          (const float*)d_in[10], (const float*)d_in[13]};
    const float* g[5]  = {(const float*)d_in[2], (const float*)d_in[5], (const float*)d_in[8],
                          (const float*)d_in[11], (const float*)d_in[14]};
    const float* bb[5] = {(const float*)d_in[3], (const float*)d_in[6], (const float*)d_in[9],
                          (const float*)d_in[12], (const float*)d_in[15]};

    char* ws = (char*)d_ws;
    size_t off = 0;
    auto carve = [&](size_t bytes) -> char* {
        char* p = ws + off;
        off = (off + bytes + 255) & ~(size_t)255;
        return p;
    };
    float* XC   = (float*)carve((size_t)BATCH * XC_CH * NPTS * 4);   // 10.6 MB
    float* DIST = (float*)carve((size_t)BATCH * NPTS * NPTS * 4);    // 32 MB (also Y5)
    int*   IDX  = (int*)  carve((size_t)BATCH * NPTS * KNN * 4);     // 0.66 MB
    float* WSTK = (float*)carve((size_t)256 * 195 * 4);              // 0.2 MB
    float* YA   = (float*)carve((size_t)BATCH * 256 * NPTS * 4);     // 8 MB
    float* MBUF = (float*)carve((size_t)BATCH * 128 * NPTS * 4);     // 4 MB
    float* XX   = (float*)carve((size_t)BATCH * NPTS * 4);
    float* SUMS = (float*)carve(1024 * 4);
    float* SCSH = (float*)carve(1024 * 4);

    k_copy_x<<<(BATCH * 3 * NPTS) / 256, 256, 0, stream>>>(x, XC);

    struct Stage { int knnOff, Cknn, ci, co, outOff; };
    const Stage st[4] = {
        {0,   3,  3,   64,  3},
        {3,   64, 67,  64,  67},
        {67,  64, 131, 64,  131},
        {131, 64, 195, 128, 195},
    };
    const float invCntEdge = 1.f / (float)(BATCH * NPTS * KNN);

    for (int s = 0; s < 4; ++s) {
        const Stage& S = st[s];
        k_xx<<<(BATCH * NPTS) / 256, 256, 0, stream>>>(XC, XX, S.knnOff, S.Cknn);
        k_gram_dist<<<dim3(NPTS / 64, NPTS / 64, BATCH), 128, 0, stream>>>(
            XC, XX, DIST, S.knnOff, S.Cknn);
        k_topk<<<(BATCH * NPTS) / 256, 256, 0, stream>>>(DIST, IDX);
        int nstk = 2 * S.co * S.ci;
        k_stack_w<<<(nstk + 255) / 256, 256, 0, stream>>>(w[s], WSTK, S.co, S.ci);
        k_gemm<<<dim3(NPTS / 64, (2 * S.co) / 64, BATCH), 128, 0, stream>>>(
            WSTK, XC, YA, 2 * S.co, S.ci);
        hipMemsetAsync(SUMS, 0, 1024 * 4, stream);
        k_edge_reduce<<<dim3((BATCH * NPTS) / 256, S.co), 256, 0, stream>>>(
            YA, IDX, MBUF, SUMS, S.co);
        k_bn_params<<<1, 256, 0, stream>>>(SUMS, g[s], bb[s], SCSH, S.co, invCntEdge);
        k_apply_feat<<<(BATCH * S.co * NPTS) / 256, 256, 0, stream>>>(
            MBUF, SCSH, XC, S.co, S.outOff);
    }

    // ---- stage 5: plain 1x1 conv (512 x 323) + BN + ReLU -> d_out ----
    float* Y5 = DIST;  // dist buffer dead; 16 MB needed, 32 MB available
    k_gemm<<<dim3(NPTS / 64, 512 / 64, BATCH), 128, 0, stream>>>(w[4], XC, Y5, 512, 323);
    hipMemsetAsync(SUMS, 0, 1024 * 4, stream);
    k_stats_plain<<<dim3((BATCH * NPTS) / 256, 512), 256, 0, stream>>>(Y5, SUMS, 512);
    k_bn_params<<<2, 256, 0, stream>>>(SUMS, g[4], bb[4], SCSH, 512,
                                       1.f / (float)(BATCH * NPTS));
    k_apply_out<<<(BATCH * 512 * NPTS) / 256, 256, 0, stream>>>(Y5, SCSH, (float*)d_out);
}